// OTVCA_59579786330139
// MI455X (gfx1250) — compile-verified
//
#include <hip/hip_runtime.h>

// ---------------------------------------------------------------------------
// OTVCA for MI455X (gfx1250): fp32 WMMA (V_WMMA_F32_16X16X4_F32) for all
// matmuls, LDS-staged B operands, branchless masked padding loads,
// LDS-resident subspace iteration + Newton-Schulz polar factor,
// float4-vectorized TV split-Bregman stencils.
// ---------------------------------------------------------------------------

typedef __attribute__((ext_vector_type(2))) float v2f;
typedef __attribute__((ext_vector_type(8))) float v8f;

#define NR 400
#define NC 400
#define NP (NR*NC)          // 160000 pixels
#define P  200              // bands
#define PP 208              // bands padded to 13*16
#define F  8                // features
#define FP 16               // features padded to 16 (WMMA N)
#define NPF (NP*F)

#define WEIGHT 100.0f       // 1/LAM
#define TVLAM  200.0f       // 2*weight
#define INVNORMC (1.0f/900.0f)  // 1/(weight + 4*lam)

// workspace layout (float offsets, all 128B aligned)
#define OFF_G    0                       // 208*208
#define OFF_V    43264                   // 208*16
#define OFF_M    46592                   // 208*16
#define OFF_PC   49920                   // NP*8
#define OFF_U1   1329920                 // NP*8
#define OFF_U2   2609920                 // NP*8
#define OFF_DX   3889920                 // NP*8
#define OFF_DY   5169920                 // NP*8
#define OFF_BX   6449920                 // NP*8
#define OFF_BY   7729920                 // NP*8  (end = 9009920 floats)

// ---------------------------------------------------------------------------
__global__ void otvca_zero(float* __restrict__ p, int n) {
    int i = blockIdx.x * blockDim.x + threadIdx.x;
    if (i < n) p[i] = 0.0f;
}

__global__ void otvca_copy(const float* __restrict__ s, float* __restrict__ d, int n) {
    int i = blockIdx.x * blockDim.x + threadIdx.x;
    if (i < n) d[i] = s[i];
}

// deterministic pseudo-random init for V (200x8 in a 208x16 padded buffer)
__global__ void otvca_init_v(float* __restrict__ V) {
    int i = blockIdx.x * blockDim.x + threadIdx.x;
    if (i >= PP * FP) return;
    int row = i >> 4, col = i & 15;
    float val = 0.0f;
    if (row < P && col < F) {
        unsigned h = (unsigned)(row * 31u + col * 131u + 7u);
        h ^= h >> 13; h *= 0x9E3779B1u; h ^= h >> 16;
        val = ((float)(h & 0xFFFFu) * (1.0f / 65536.0f)) - 0.5f;
    }
    V[i] = val;
}

// ---------------------------------------------------------------------------
// G += X^T X  : A[m][k] = X[pix_k][band_m], B[k][n] = X[pix_k][band_n]
// grid = (chunks, 169), block = 32 (one wave per output tile slice).
// Padding handled branchlessly: clamp address + multiply by 0/1 mask, so the
// inner loop is unconditional global_load_b32 stream + WMMA (no exec churn).
__global__ void otvca_gram_wmma(const float* __restrict__ X, float* __restrict__ G,
                                int pixPerWave) {
    const int lane = threadIdx.x & 31;
    const int half = lane >> 4;
    const int l15  = lane & 15;
    const int tile = blockIdx.y;            // 0..168
    const int tm = (tile % 13) * 16;
    const int tn = (tile / 13) * 16;
    const int m = tm + l15;
    const int n = tn + l15;
    const int   mc = (m < P) ? m : 0;       // clamped address
    const float ma = (m < P) ? 1.0f : 0.0f; // pad mask
    const int   nc2 = (n < P) ? n : 0;
    const float mb  = (n < P) ? 1.0f : 0.0f;
    const int pix0 = blockIdx.x * pixPerWave;
    v8f c = {};
    for (int k = 0; k < pixPerWave; k += 4) {
        const int pk = pix0 + k + 2 * half;
        const float* x0 = X + (size_t)pk * P;
        __builtin_prefetch(x0 + 8 * (size_t)P, 0, 1);
        v2f a, b;
        a.x = x0[mc]      * ma;
        a.y = x0[P + mc]  * ma;
        b.x = x0[nc2]     * mb;
        b.y = x0[P + nc2] * mb;
        c = __builtin_amdgcn_wmma_f32_16x16x4_f32(false, a, false, b, (short)0, c,
                                                  false, false);
    }
#pragma unroll
    for (int r = 0; r < 8; ++r) {
        const int gm = tm + r + 8 * half;
        unsafeAtomicAdd(&G[gm * PP + n], c[r]);
    }
}

// ---------------------------------------------------------------------------
// one round of subspace iteration: W = G*V ; V = MGS-orth(W). single workgroup.
__global__ void __launch_bounds__(256)
otvca_power_step(const float* __restrict__ G, float* __restrict__ V) {
    __shared__ float Vl[PP * FP];
    __shared__ float Wl[PP * FP];
    __shared__ float red[256];
    const int t = threadIdx.x;
    for (int i = t; i < PP * FP; i += 256) Vl[i] = V[i];
    __syncthreads();
    for (int e = t; e < PP * FP; e += 256) {
        const int i = e >> 4, j = e & 15;
        float s = 0.0f;
        for (int k = 0; k < PP; ++k) s += G[i * PP + k] * Vl[k * FP + j];
        Wl[e] = s;
    }
    __syncthreads();
    for (int j = 0; j < F; ++j) {
        for (int i = 0; i < j; ++i) {
            float part = 0.0f;
            for (int r = t; r < PP; r += 256) part += Vl[r * FP + i] * Wl[r * FP + j];
            red[t] = part; __syncthreads();
            for (int s = 128; s > 0; s >>= 1) { if (t < s) red[t] += red[t + s]; __syncthreads(); }
            const float rij = red[0]; __syncthreads();
            for (int r = t; r < PP; r += 256) Wl[r * FP + j] -= rij * Vl[r * FP + i];
            __syncthreads();
        }
        float part = 0.0f;
        for (int r = t; r < PP; r += 256) { const float w = Wl[r * FP + j]; part += w * w; }
        red[t] = part; __syncthreads();
        for (int s = 128; s > 0; s >>= 1) { if (t < s) red[t] += red[t + s]; __syncthreads(); }
        const float inv = rsqrtf(red[0] + 1e-30f); __syncthreads();
        for (int r = t; r < PP; r += 256) Vl[r * FP + j] = Wl[r * FP + j] * inv;
        __syncthreads();
    }
    for (int i = t; i < PP * FP; i += 256)
        V[i] = ((i & 15) < F) ? Vl[i] : 0.0f;
}

// ---------------------------------------------------------------------------
// PC = X @ V : per wave one 16-pixel tile, K accumulated 200/4 = 50 WMMAs.
// V staged in LDS once per block; B operand fed from ds_load (co-executes
// with WMMA), VMEM reserved for the streaming X reads.
__global__ void __launch_bounds__(256)
otvca_proj_wmma(const float* __restrict__ X, const float* __restrict__ V,
                float* __restrict__ PC) {
    __shared__ float Vl[PP * FP];
    for (int i = threadIdx.x; i < PP * FP; i += 256) Vl[i] = V[i];
    __syncthreads();
    const int lane = threadIdx.x & 31;
    const int wave = threadIdx.x >> 5;
    const int half = lane >> 4;
    const int l15  = lane & 15;
    const int tile = blockIdx.x * 8 + wave;         // 0..9999
    const int row  = tile * 16 + l15;
    const float* xr = X + (size_t)row * P + 2 * half;
    const float* vb = Vl + 2 * half * FP + l15;
    v8f c = {};
    for (int kb = 0; kb < P; kb += 4) {
        __builtin_prefetch(xr + kb + 64, 0, 1);
        v2f a = *(const v2f*)(xr + kb);
        v2f b;
        b.x = vb[kb * FP];
        b.y = vb[kb * FP + FP];
        c = __builtin_amdgcn_wmma_f32_16x16x4_f32(false, a, false, b, (short)0, c,
                                                  false, false);
    }
    if (l15 < F) {
#pragma unroll
        for (int r = 0; r < 8; ++r)
            PC[(size_t)(tile * 16 + r + 8 * half) * F + l15] = c[r];
    }
}

// ---------------------------------------------------------------------------
// M += X^T @ FE : A[m][k] = X[pix_k][band_m], B[k][n] = FE[pix_k][n]
// branchless pad masks as in gram.
__global__ void otvca_corr_wmma(const float* __restrict__ X, const float* __restrict__ FE,
                                float* __restrict__ M, int pixPerWave) {
    const int lane = threadIdx.x & 31;
    const int half = lane >> 4;
    const int l15  = lane & 15;
    const int bt   = blockIdx.y;                    // 0..12
    const int band = bt * 16 + l15;
    const int   bc = (band < P) ? band : 0;
    const float ma = (band < P) ? 1.0f : 0.0f;
    const int   fc = (l15 < F) ? l15 : 0;
    const float mb = (l15 < F) ? 1.0f : 0.0f;
    const int pix0 = blockIdx.x * pixPerWave;
    v8f c = {};
    for (int k = 0; k < pixPerWave; k += 4) {
        const int pk = pix0 + k + 2 * half;
        const float* x0 = X  + (size_t)pk * P;
        const float* f0 = FE + (size_t)pk * F;
        __builtin_prefetch(x0 + 8 * (size_t)P, 0, 1);
        v2f a, b;
        a.x = x0[bc]     * ma;
        a.y = x0[P + bc] * ma;
        b.x = f0[fc]     * mb;
        b.y = f0[F + fc] * mb;
        c = __builtin_amdgcn_wmma_f32_16x16x4_f32(false, a, false, b, (short)0, c,
                                                  false, false);
    }
#pragma unroll
    for (int r = 0; r < 8; ++r)
        unsafeAtomicAdd(&M[(bt * 16 + r + 8 * half) * FP + l15], c[r]);
}

// ---------------------------------------------------------------------------
// Newton-Schulz polar factor of M (200x8):  V = M (M^T M)^{-1/2}. one workgroup.
__global__ void __launch_bounds__(256)
otvca_polar(const float* __restrict__ Mmat, float* __restrict__ V) {
    __shared__ float Y[P * F];
    __shared__ float A8[F * F];
    __shared__ float red[256];
    const int t = threadIdx.x;
    float part = 0.0f;
    for (int e = t; e < P * F; e += 256) {
        const float v = Mmat[(e / F) * FP + (e % F)];
        Y[e] = v; part += v * v;
    }
    red[t] = part; __syncthreads();
    for (int s = 128; s > 0; s >>= 1) { if (t < s) red[t] += red[t + s]; __syncthreads(); }
    const float inv = rsqrtf(red[0] + 1e-30f);
    __syncthreads();
    for (int e = t; e < P * F; e += 256) Y[e] *= inv;
    __syncthreads();
    for (int it = 0; it < 14; ++it) {
        if (t < F * F) A8[t] = 0.0f;
        __syncthreads();
        for (int r = t; r < P; r += 256) {
#pragma unroll
            for (int i = 0; i < F; ++i)
#pragma unroll
                for (int j = 0; j < F; ++j)
                    atomicAdd(&A8[i * F + j], Y[r * F + i] * Y[r * F + j]);
        }
        __syncthreads();
        for (int r = t; r < P; r += 256) {
            float yr[F];
#pragma unroll
            for (int j = 0; j < F; ++j) yr[j] = Y[r * F + j];
#pragma unroll
            for (int j = 0; j < F; ++j) {
                float acc = 0.0f;
#pragma unroll
                for (int k = 0; k < F; ++k) acc += yr[k] * A8[k * F + j];
                Y[r * F + j] = 1.5f * yr[j] - 0.5f * acc;
            }
        }
        __syncthreads();
    }
    for (int e = t; e < PP * FP; e += 256) {
        const int r = e >> 4, c = e & 15;
        V[e] = (r < P && c < F) ? Y[r * F + c] : 0.0f;
    }
}

// ---------------------------------------------------------------------------
// TV split-Bregman: u-update (Jacobi). one thread = one pixel x 4 channels.
__global__ void otvca_tv_u(const float4* __restrict__ u, const float4* __restrict__ img,
                           const float4* __restrict__ dx, const float4* __restrict__ dy,
                           const float4* __restrict__ bx, const float4* __restrict__ by,
                           float4* __restrict__ un) {
    const int t = blockIdx.x * blockDim.x + threadIdx.x;
    if (t >= NP * 2) return;
    const int pix = t >> 1, g = t & 1;
    const int r = pix / NC, c = pix % NC;
    const int i  = pix * 2 + g;
    const int iN = (r > 0      ? pix - NC : pix) * 2 + g;
    const int iS = (r < NR - 1 ? pix + NC : pix) * 2 + g;
    const int iW = (c > 0      ? pix - 1  : pix) * 2 + g;
    const int iE = (c < NC - 1 ? pix + 1  : pix) * 2 + g;
    const float4 uN = u[iN], uS = u[iS], uW = u[iW], uE = u[iE];
    const float4 dxc = dx[i], dyc = dy[i], bxc = bx[i], byc = by[i];
    // zero-at-boundary aux reads, branchless via mask
    const float wm = (c > 0) ? 1.0f : 0.0f;
    const float nm = (r > 0) ? 1.0f : 0.0f;
    const float4 dxw0 = dx[iW], dyn0 = dy[iN], bxw0 = bx[iW], byn0 = by[iN];
    const float4 im = img[i];
    float4 o;
#define TVU_C(f) o.f = (TVLAM * (uN.f + uS.f + uE.f + uW.f \
        + wm * (dxw0.f - bxw0.f) + nm * (dyn0.f - byn0.f) \
        - dxc.f - dyc.f + bxc.f + byc.f) \
        + WEIGHT * im.f) * INVNORMC;
    TVU_C(x) TVU_C(y) TVU_C(z) TVU_C(w)
#undef TVU_C
    un[i] = o;
}

// TV split-Bregman: isotropic shrink + Bregman update (in place on d/b).
__global__ void otvca_tv_db(const float4* __restrict__ u,
                            float4* __restrict__ dx, float4* __restrict__ dy,
                            float4* __restrict__ bx, float4* __restrict__ by) {
    const int t = blockIdx.x * blockDim.x + threadIdx.x;
    if (t >= NP * 2) return;
    const int pix = t >> 1, g = t & 1;
    const int r = pix / NC, c = pix % NC;
    const int i  = pix * 2 + g;
    const int iS = (r < NR - 1 ? pix + NC : pix) * 2 + g;
    const int iE = (c < NC - 1 ? pix + 1  : pix) * 2 + g;
    const float4 uc = u[i], uS = u[iS], uE = u[iE];
    const float4 bxc = bx[i], byc = by[i];
    float4 odx, ody, obx, oby;
#define TVDB_C(f) { \
        const float ux = uE.f - uc.f; \
        const float uy = uS.f - uc.f; \
        const float tx = ux + bxc.f; \
        const float ty = uy + byc.f; \
        const float s  = sqrtf(fmaxf(tx * tx + ty * ty, 1e-12f)); \
        const float sc = s * TVLAM / (s * TVLAM + 1.0f); \
        const float dxn = sc * tx; \
        const float dyn3 = sc * ty; \
        odx.f = dxn; ody.f = dyn3; \
        obx.f = bxc.f + ux - dxn; oby.f = byc.f + uy - dyn3; }
    TVDB_C(x) TVDB_C(y) TVDB_C(z) TVDB_C(w)
#undef TVDB_C
    dx[i] = odx; dy[i] = ody; bx[i] = obx; by[i] = oby;
}

// ---------------------------------------------------------------------------
// out = FE @ V^T : A = 16 pixels x 4 features (K=8 -> 2 WMMAs), B[k][n]=V[band][k]
// V staged in LDS (padded rows are zero, so only stores need the band guard).
__global__ void __launch_bounds__(256)
otvca_recon_wmma(const float* __restrict__ FE, const float* __restrict__ V,
                 float* __restrict__ out) {
    __shared__ float Vl[PP * FP];
    for (int i = threadIdx.x; i < PP * FP; i += 256) Vl[i] = V[i];
    __syncthreads();
    const int lane = threadIdx.x & 31;
    const int wave = threadIdx.x >> 5;
    const int half = lane >> 4;
    const int l15  = lane & 15;
    const int mt   = blockIdx.x * 8 + wave;         // 0..9999
    const int nt   = blockIdx.y;                    // 0..12
    const int band = nt * 16 + l15;
    v8f c = {};
#pragma unroll
    for (int kb = 0; kb < F; kb += 4) {
        v2f a = *(const v2f*)(FE + (size_t)(mt * 16 + l15) * F + kb + 2 * half);
        v2f b;
        b.x = Vl[band * FP + kb + 2 * half];
        b.y = Vl[band * FP + kb + 2 * half + 1];
        c = __builtin_amdgcn_wmma_f32_16x16x4_f32(false, a, false, b, (short)0, c,
                                                  false, false);
    }
    if (band < P) {
#pragma unroll
        for (int r = 0; r < 8; ++r)
            out[(size_t)(mt * 16 + r + 8 * half) * P + band] = c[r];
    }
}

// ---------------------------------------------------------------------------
extern "C" void kernel_launch(void* const* d_in, const int* in_sizes, int n_in,
                              void* d_out, int out_size, void* d_ws, size_t ws_size,
                              hipStream_t stream) {
    const float* X = (const float*)d_in[0];   // (400,400,200) fp32
    // d_in[1]=features(=8), d_in[2]=num_itt(=10) are compile-time constants here.
    float* ws = (float*)d_ws;
    float* G  = ws + OFF_G;
    float* V  = ws + OFF_V;
    float* Mm = ws + OFF_M;
    float* PC = ws + OFF_PC;
    float* U1 = ws + OFF_U1;
    float* U2 = ws + OFF_U2;
    float* DX = ws + OFF_DX;
    float* DY = ws + OFF_DY;
    float* BX = ws + OFF_BX;
    float* BY = ws + OFF_BY;
    float* OUT = (float*)d_out;

    // --- initial subspace: G = X^T X, then 30 rounds of orthogonal iteration
    otvca_zero<<<(PP * PP + 255) / 256, 256, 0, stream>>>(G, PP * PP);
    otvca_gram_wmma<<<dim3(50, 169), 32, 0, stream>>>(X, G, NP / 50);
    otvca_init_v<<<(PP * FP + 255) / 256, 256, 0, stream>>>(V);
    for (int i = 0; i < 30; ++i)
        otvca_power_step<<<1, 256, 0, stream>>>(G, V);

    // --- outer OTVCA loop
    for (int outer = 0; outer < 10; ++outer) {
        otvca_proj_wmma<<<1250, 256, 0, stream>>>(X, V, PC);
        otvca_zero<<<(4 * NPF + 255) / 256, 256, 0, stream>>>(DX, 4 * NPF);
        const float* usrc = PC;
        float* udst = U1;
        for (int inner = 0; inner < 10; ++inner) {
            otvca_tv_u<<<(NP * 2 + 255) / 256, 256, 0, stream>>>(
                (const float4*)usrc, (const float4*)PC,
                (const float4*)DX, (const float4*)DY,
                (const float4*)BX, (const float4*)BY, (float4*)udst);
            otvca_tv_db<<<(NP * 2 + 255) / 256, 256, 0, stream>>>(
                (const float4*)udst, (float4*)DX, (float4*)DY,
                (float4*)BX, (float4*)BY);
            usrc = udst;
            udst = (udst == U1) ? U2 : U1;
        }
        // FE = usrc (== U2 after 10 sweeps)
        otvca_zero<<<(PP * FP + 255) / 256, 256, 0, stream>>>(Mm, PP * FP);
        otvca_corr_wmma<<<dim3(125, 13), 32, 0, stream>>>(X, usrc, Mm, NP / 125);
        otvca_polar<<<1, 256, 0, stream>>>(Mm, V);
    }

    // --- outputs: denoised (NP*P) then fe (NP*F)
    const float* FE = U2;
    otvca_recon_wmma<<<dim3(1250, 13), 256, 0, stream>>>(FE, V, OUT);
    otvca_copy<<<(NPF + 255) / 256, 256, 0, stream>>>(FE, OUT + (size_t)NP * P, NPF);
}